// FwpLmVariantC_33371895890548
// MI455X (gfx1250) — compile-verified
//
#include <hip/hip_runtime.h>
#include <hip/hip_bf16.h>
#include <math.h>
#include <stdint.h>

#define D_DIM 512
#define H_DIM 20
#define V_DIM 32000
#define T_DIM 1024
#define K_DIM 512           // both GEMMs have K = D = 512
#define A_PITCH 516         // LDS row pitch (floats): 4-bank shift/row -> conflict-free

typedef __attribute__((ext_vector_type(2))) float v2f;
typedef __attribute__((ext_vector_type(8))) float v8f;
typedef __attribute__((ext_vector_type(4))) int   v4i;

#define GLOBAL_AS __attribute__((address_space(1)))
#define LDS_AS    __attribute__((address_space(3)))

// ---- CDNA5 async global->LDS copy (ASYNCcnt-tracked), with safe fallback ----
__device__ __forceinline__ void async_copy_b128(const void* g, void* l) {
#if defined(__has_builtin) && __has_builtin(__builtin_amdgcn_global_load_async_to_lds_b128)
    __builtin_amdgcn_global_load_async_to_lds_b128(
        (GLOBAL_AS v4i*)(uintptr_t)g,
        (LDS_AS v4i*)(uint32_t)(uintptr_t)l,   // flat-shared low 32 bits == LDS offset
        0, 0);
#else
    *(float4*)l = *(const float4*)g;
#endif
}

template <int N>
__device__ __forceinline__ void wait_async() {
#if defined(__has_builtin) && __has_builtin(__builtin_amdgcn_s_wait_asynccnt)
    __builtin_amdgcn_s_wait_asynccnt(N);
#else
    asm volatile("s_wait_asynccnt %0" :: "i"(N) : "memory");
#endif
}

// ---------------------------------------------------------------------------
// 1) Embedding gather: xs[t,:] = embed[tokens[t],:]
// ---------------------------------------------------------------------------
__global__ void gather_embed_kernel(const int* __restrict__ tokens,
                                    const float* __restrict__ embed,
                                    float* __restrict__ xs) {
    const int t = blockIdx.x;
    const int row = tokens[t];
    const float4* src = (const float4*)(embed + (size_t)row * D_DIM);
    float4* dst = (float4*)(xs + (size_t)t * D_DIM);
    dst[threadIdx.x] = src[threadIdx.x];   // 128 threads * float4 = 512 floats
}

// ---------------------------------------------------------------------------
// 2) FP32 WMMA GEMM: C[M,N] = A[M,K=512] * B[K,N] (+ bias[n]); row-major.
//    Block = 4 waves = 16(M) x 256(N) tile. The 16x512 A tile (32 KB) is
//    staged once into LDS via async b128 copies, then each wave feeds its
//    V_WMMA_F32_16X16X4_F32 A-fragments from conflict-free ds_load_b64.
//    B (streaming, L2-resident) loaded coalesced + branch-free prefetch ahead.
// ---------------------------------------------------------------------------
template <bool HAS_BIAS>
__global__ void gemm_wmma_f32_kernel(const float* __restrict__ A,
                                     const float* __restrict__ B,
                                     const float* __restrict__ bias,
                                     float* __restrict__ C,
                                     int M, int N) {
    __shared__ __align__(16) float sA[16 * A_PITCH];

    const int tid  = threadIdx.x;
    const int wave = tid >> 5;
    const int lane = tid & 31;
    const int half = lane >> 4;      // 0: K pair {0,1} / rows 0-7 ; 1: {2,3} / rows 8-15
    const int l16  = lane & 15;

    const int mrow0 = blockIdx.x * 16;
    const int ncol0 = blockIdx.y * 256 + wave * 64;

    // ---- stage A tile: 16 rows x 512 floats = 2048 float4 chunks ----
    for (int c = tid; c < 16 * (K_DIM / 4); c += 128) {
        const int row = c >> 7;           // / (K_DIM/4)
        const int ch  = c & 127;
        async_copy_b128(A + (size_t)(mrow0 + row) * K_DIM + ch * 4,
                        &sA[row * A_PITCH + ch * 4]);
    }
    wait_async<0>();
    __syncthreads();

    v8f acc0 = {}, acc1 = {}, acc2 = {}, acc3 = {};
    const int aoff = l16 * A_PITCH + 2 * half;
    const int nbase = ncol0 + l16;

#pragma unroll 4
    for (int k0 = 0; k0 < K_DIM; k0 += 4) {
        v2f a;
        a.x = sA[aoff + k0];
        a.y = sA[aoff + k0 + 1];

        const float* Bk = B + (size_t)(k0 + 2 * half) * N;
        // branch-free stream-ahead prefetch (speculative; safe past the end)
        __builtin_prefetch(Bk + (size_t)32 * N + nbase, 0, 1);

        v2f b0, b1, b2, b3;
        b0.x = Bk[nbase +  0]; b0.y = Bk[N + nbase +  0];
        b1.x = Bk[nbase + 16]; b1.y = Bk[N + nbase + 16];
        b2.x = Bk[nbase + 32]; b2.y = Bk[N + nbase + 32];
        b3.x = Bk[nbase + 48]; b3.y = Bk[N + nbase + 48];

        acc0 = __builtin_amdgcn_wmma_f32_16x16x4_f32(false, a, false, b0, (short)0, acc0, false, false);
        acc1 = __builtin_amdgcn_wmma_f32_16x16x4_f32(false, a, false, b1, (short)0, acc1, false, false);
        acc2 = __builtin_amdgcn_wmma_f32_16x16x4_f32(false, a, false, b2, (short)0, acc2, false, false);
        acc3 = __builtin_amdgcn_wmma_f32_16x16x4_f32(false, a, false, b3, (short)0, acc3, false, false);
    }

    // C/D layout: VGPR r -> row (r + 8*half), lanes l16 -> column
    v8f accs[4] = {acc0, acc1, acc2, acc3};
#pragma unroll
    for (int nt = 0; nt < 4; ++nt) {
        const int n = ncol0 + nt * 16 + l16;
        const float bv = HAS_BIAS ? bias[n] : 0.0f;
#pragma unroll
        for (int r = 0; r < 8; ++r) {
            const int m = mrow0 + r + 8 * half;
            C[(size_t)m * N + n] = accs[nt][r] + bv;
        }
    }
}

// ---------------------------------------------------------------------------
// 3) Fast-weight scan. Columns of fw are fully independent:
//       fw[:,j] <- tanh(fw[:,j] + eta*z1[t][j]*z0[t][:]);  y[t][j] = dot(x[t], fw[:,j])
//    Block = 256 threads = 8 columns x 32 lanes; each lane keeps 16 fw rows
//    in VGPRs for 1024 steps. z0/x rows double-buffered via async b128 so the
//    next step's L2 broadcast overlaps this step's tanh/FMA chain.
// ---------------------------------------------------------------------------
__global__ void fw_scan_kernel(const float* __restrict__ xs,
                               const float* __restrict__ z0,
                               const float* __restrict__ z1,
                               const float* __restrict__ fw_init,
                               const float* __restrict__ eta_p,
                               float* __restrict__ ys) {
    __shared__ __align__(16) float s_z0[2][D_DIM];
    __shared__ __align__(16) float s_x[2][D_DIM];
    __shared__ float s_z1[8];

    const int tid  = threadIdx.x;
    const int seg  = tid & 31;      // lane within column's wave: rows seg*16..seg*16+15
    const int cidx = tid >> 5;      // 0..7 column within block
    const int col  = blockIdx.x * 8 + cidx;
    const float eta = *eta_p;

    float fw[16];
#pragma unroll
    for (int i = 0; i < 16; ++i)
        fw[i] = fw_init[(size_t)(seg * 16 + i) * D_DIM + col];

    // prologue: stage step 0 into buffer 0 (1 async b128 per thread)
    if (tid < 128) async_copy_b128(z0 + tid * 4, &s_z0[0][tid * 4]);
    else           async_copy_b128(xs + (tid - 128) * 4, &s_x[0][(tid - 128) * 4]);

    for (int t = 0; t < T_DIM; ++t) {
        const int cur = t & 1, nxt = cur ^ 1;
        // issue step t+1 (outstanding per wave: 2); async loads complete in
        // order, so waiting for <=1 guarantees buffer `cur` has landed.
        if (t + 1 < T_DIM) {
            const float* z0n = z0 + (size_t)(t + 1) * D_DIM;
            const float* xn  = xs + (size_t)(t + 1) * D_DIM;
            if (tid < 128) async_copy_b128(z0n + tid * 4, &s_z0[nxt][tid * 4]);
            else           async_copy_b128(xn + (tid - 128) * 4, &s_x[nxt][(tid - 128) * 4]);
            wait_async<1>();
        } else {
            wait_async<0>();
        }
        if (tid < 8) s_z1[tid] = z1[(size_t)t * D_DIM + blockIdx.x * 8 + tid];
        __syncthreads();

        const float ez1 = eta * s_z1[cidx];
        float acc = 0.0f;
#pragma unroll
        for (int i = 0; i < 16; ++i) {
            const int r = seg * 16 + i;
            float f = tanhf(fw[i] + ez1 * s_z0[cur][r]);
            fw[i] = f;
            acc += s_x[cur][r] * f;
        }
        // full-wave reduction over 32 lanes (one column per wave)
        for (int m = 16; m > 0; m >>= 1)
            acc += __shfl_xor(acc, m, 32);
        if (seg == 0) ys[(size_t)t * D_DIM + col] = acc;
        __syncthreads();
    }
}

// ---------------------------------------------------------------------------
// 4) Tiny MLP + mix: mixed = 0.5*(selu(y@aw+ab)@bw+bb) + 0.5*x. One block/token.
// ---------------------------------------------------------------------------
__global__ void mlp_mix_kernel(const float* __restrict__ ys,
                               const float* __restrict__ xs,
                               const float* __restrict__ aw,   // [D,H]
                               const float* __restrict__ ab,   // [H]
                               const float* __restrict__ bw,   // [H,D]
                               const float* __restrict__ bb,   // [D]
                               float* __restrict__ mixed) {
    __shared__ float yb[D_DIM];
    __shared__ float hb[H_DIM];
    const int t = blockIdx.x;
    const int tid = threadIdx.x;

    yb[tid]       = ys[(size_t)t * D_DIM + tid];
    yb[tid + 256] = ys[(size_t)t * D_DIM + 256 + tid];
    __syncthreads();

    if (tid < H_DIM) {
        float u = ab[tid];
        for (int i = 0; i < D_DIM; ++i)
            u += yb[i] * aw[i * H_DIM + tid];
        const float scale = 1.0507009873554805f;
        const float alpha = 1.6732632423543772f;
        hb[tid] = (u > 0.0f) ? (scale * u) : (scale * alpha * (expf(u) - 1.0f));
    }
    __syncthreads();

    for (int d = tid; d < D_DIM; d += 256) {
        float s = bb[d];
#pragma unroll
        for (int j = 0; j < H_DIM; ++j)
            s += hb[j] * bw[j * D_DIM + d];
        mixed[(size_t)t * D_DIM + d] = 0.5f * s + 0.5f * xs[(size_t)t * D_DIM + d];
    }
}

// ---------------------------------------------------------------------------
// 5) Per-row online log-softmax loss: rowloss[t] = -(logit[tgt]-max-log(sumexp))
//    Logits are a 131 MB read-once stream -> nontemporal loads keep L2 for
//    head_w.
// ---------------------------------------------------------------------------
__global__ void softmax_loss_kernel(const float* __restrict__ logits,
                                    const int* __restrict__ targets,
                                    float* __restrict__ rowloss) {
    const int t = blockIdx.x;
    const float* row = logits + (size_t)t * V_DIM;

    float m = -3.4e38f, s = 0.0f;
    for (int n = threadIdx.x; n < V_DIM; n += 256) {
        const float v = __builtin_nontemporal_load(&row[n]);
        const float mn = fmaxf(m, v);
        s = s * expf(m - mn) + expf(v - mn);
        m = mn;
    }
    // wave reduce (wave32)
    for (int k = 16; k > 0; k >>= 1) {
        const float m2 = __shfl_xor(m, k, 32);
        const float s2 = __shfl_xor(s, k, 32);
        const float mn = fmaxf(m, m2);
        s = s * expf(m - mn) + s2 * expf(m2 - mn);
        m = mn;
    }
    __shared__ float sm[8], ss[8];
    const int wave = threadIdx.x >> 5, lane = threadIdx.x & 31;
    if (lane == 0) { sm[wave] = m; ss[wave] = s; }
    __syncthreads();
    if (threadIdx.x == 0) {
        m = sm[0]; s = ss[0];
        for (int w = 1; w < 8; ++w) {
            const float mn = fmaxf(m, sm[w]);
            s = s * expf(m - mn) + ss[w] * expf(sm[w] - mn);
            m = mn;
        }
        const float lt = row[targets[t]];
        rowloss[t] = -(lt - m - logf(s));
    }
}

__global__ void loss_reduce_kernel(const float* __restrict__ rowloss,
                                   float* __restrict__ out_loss) {
    __shared__ float buf[256];
    float s = 0.0f;
    for (int i = threadIdx.x; i < T_DIM; i += 256) s += rowloss[i];
    buf[threadIdx.x] = s;
    __syncthreads();
    for (int k = 128; k > 0; k >>= 1) {
        if (threadIdx.x < k) buf[threadIdx.x] += buf[threadIdx.x + k];
        __syncthreads();
    }
    if (threadIdx.x == 0) out_loss[0] = buf[0] * (1.0f / (float)T_DIM);
}

// ---------------------------------------------------------------------------
// Launch
// ---------------------------------------------------------------------------
extern "C" void kernel_launch(void* const* d_in, const int* in_sizes, int n_in,
                              void* d_out, int out_size, void* d_ws, size_t ws_size,
                              hipStream_t stream) {
    const int*   tokens  = (const int*)  d_in[0];
    const int*   targets = (const int*)  d_in[1];
    const float* embed   = (const float*)d_in[2];
    const float* fw_init = (const float*)d_in[3];
    const float* eta     = (const float*)d_in[4];
    const float* wa      = (const float*)d_in[5];
    const float* wb      = (const float*)d_in[6];
    const float* mlp_aw  = (const float*)d_in[7];
    const float* mlp_ab  = (const float*)d_in[8];
    const float* mlp_bw  = (const float*)d_in[9];
    const float* mlp_bb  = (const float*)d_in[10];
    const float* head_w  = (const float*)d_in[11];
    const float* head_b  = (const float*)d_in[12];

    float* logits = (float*)d_out;                       // [T, V]
    float* loss   = (float*)d_out + (size_t)T_DIM * V_DIM;

    const size_t TD = (size_t)T_DIM * D_DIM;
    float* w      = (float*)d_ws;
    float* xs     = w;            // [T,D]
    float* z0     = xs + TD;      // [T,D]
    float* z1     = z0 + TD;      // [T,D]
    float* ysb    = z1 + TD;      // [T,D]
    float* mixed  = ysb + TD;     // [T,D]
    float* rowlo  = mixed + TD;   // [T]

    // 1) gather embeddings
    gather_embed_kernel<<<T_DIM, 128, 0, stream>>>(tokens, embed, xs);

    // 2) z0 = xs @ wa ; z1 = xs @ wb   (WMMA f32, N=512 -> grid.y = 2)
    gemm_wmma_f32_kernel<false><<<dim3(T_DIM / 16, D_DIM / 256), 128, 0, stream>>>(
        xs, wa, nullptr, z0, T_DIM, D_DIM);
    gemm_wmma_f32_kernel<false><<<dim3(T_DIM / 16, D_DIM / 256), 128, 0, stream>>>(
        xs, wb, nullptr, z1, T_DIM, D_DIM);

    // 3) column-parallel fast-weight scan (512 cols / 8 per block = 64 blocks)
    fw_scan_kernel<<<D_DIM / 8, 256, 0, stream>>>(xs, z0, z1, fw_init, eta, ysb);

    // 4) MLP + 0.5 mix
    mlp_mix_kernel<<<T_DIM, 256, 0, stream>>>(ysb, xs, mlp_aw, mlp_ab, mlp_bw, mlp_bb, mixed);

    // 5) logits = mixed @ head_w + head_b   (WMMA f32, N=32000 -> grid.y = 125)
    gemm_wmma_f32_kernel<true><<<dim3(T_DIM / 16, V_DIM / 256), 128, 0, stream>>>(
        mixed, head_w, head_b, logits, T_DIM, V_DIM);

    // 6) loss
    softmax_loss_kernel<<<T_DIM, 256, 0, stream>>>(logits, targets, rowlo);
    loss_reduce_kernel<<<1, 256, 0, stream>>>(rowlo, loss);
}